// GraphSAGE_50792283243093
// MI455X (gfx1250) — compile-verified
//
#include <hip/hip_runtime.h>

#define DD 128
#define LSTRIDE 132   // 16-row LDS tile stride in floats; 132%64=4 avoids bank conflicts

typedef __attribute__((ext_vector_type(2))) float v2f;
typedef __attribute__((ext_vector_type(8))) float v8f;

__device__ __forceinline__ void atomAddF(float* p, float v) {
  unsafeAtomicAdd(p, v);   // native global_atomic_add_f32 on gfx1250
}

// ---------------------------------------------------------------- degrees
__global__ void sage_deg_kernel(const int* __restrict__ src, const int* __restrict__ dst,
                                const float* __restrict__ ew,
                                float* deg_src, float* deg_dst, float* in_cnt, int E) {
  int e = blockIdx.x * blockDim.x + threadIdx.x;
  if (e >= E) return;
  float w = ew[e];
  atomAddF(&deg_src[src[e]], w);
  atomAddF(&deg_dst[dst[e]], w);
  atomAddF(&in_cnt[dst[e]], 1.0f);
}

__global__ void sage_invdeg_kernel(const float* __restrict__ in_cnt,
                                   float* __restrict__ inv_deg, int N) {
  int i = blockIdx.x * blockDim.x + threadIdx.x;
  if (i < N) inv_deg[i] = 1.0f / fmaxf(in_cnt[i], 1.0f);
}

__global__ void sage_edgenorm_kernel(const int* __restrict__ src, const int* __restrict__ dst,
                                     const float* __restrict__ ew,
                                     const float* __restrict__ deg_src,
                                     const float* __restrict__ deg_dst,
                                     float* __restrict__ w, int E) {
  int e = blockIdx.x * blockDim.x + threadIdx.x;
  if (e >= E) return;
  float d = fmaxf(deg_src[src[e]] * deg_dst[dst[e]], 1e-12f);
  w[e] = ew[e] / sqrtf(d);
}

// ------------------------------------------------ edge scatter (wave per edge)
__global__ void sage_aggregate_kernel(const int* __restrict__ src, const int* __restrict__ dst,
                                      const float* __restrict__ w,
                                      const float* __restrict__ h,
                                      float* __restrict__ neigh, int E) {
  int gid  = blockIdx.x * blockDim.x + threadIdx.x;
  int e    = gid >> 5;          // one wave32 per edge
  int lane = gid & 31;          // each lane: one float4 of the 128-wide row
  if (e >= E) return;
  int s  = src[e];
  int dv = dst[e];
  float ww = w[e];
  const float4* hp = (const float4*)(h + (size_t)s * DD);
  float4 v = hp[lane];
  float* np = neigh + (size_t)dv * DD + lane * 4;
  atomAddF(np + 0, v.x * ww);
  atomAddF(np + 1, v.y * ww);
  atomAddF(np + 2, v.z * ww);
  atomAddF(np + 3, v.w * ww);
}

// ------------------------------------------------ fused dual-GEMM + bias + ReLU
// out[16 rows x 128 cols per block] = h@Ws + (neigh/in_deg)@Wn + b  (optional ReLU)
__global__ __launch_bounds__(256)
void sage_gemm_kernel(const float* __restrict__ h, const float* __restrict__ neigh,
                      const float* __restrict__ inv_deg,
                      const float* __restrict__ Ws, const float* __restrict__ Wn,
                      const float* __restrict__ bias,
                      float* __restrict__ out, int N, int relu) {
  __shared__ float lh[16 * LSTRIDE];
  __shared__ float ln[16 * LSTRIDE];

  const int tid  = threadIdx.x;
  const int row0 = blockIdx.x * 16;

  // stage 16x128 tiles of h and neigh into LDS (two float4s per thread each)
  #pragma unroll
  for (int it = 0; it < 2; ++it) {
    int c   = tid + it * 256;        // float4 chunk id 0..511
    int m   = (c * 4) >> 7;          // row in tile
    int col = (c * 4) & 127;
    int gr  = row0 + m; if (gr > N - 1) gr = N - 1;
    float4 vh = *(const float4*)(h     + (size_t)gr * DD + col);
    float4 vn = *(const float4*)(neigh + (size_t)gr * DD + col);
    *(float4*)(&lh[m * LSTRIDE + col]) = vh;
    *(float4*)(&ln[m * LSTRIDE + col]) = vn;
  }
  __syncthreads();

  const int wave = tid >> 5;
  const int lane = tid & 31;
  const int g    = lane >> 4;        // K-pair group: lanes 0-15 -> K{0,1}, 16-31 -> K{2,3}
  const int mn   = lane & 15;        // A-row / B-col index
  const int col0 = wave * 16;        // this wave's N-tile

  int irow = row0 + mn; if (irow > N - 1) irow = N - 1;
  const float invd = inv_deg[irow];

  v8f acc = {};
  #pragma unroll
  for (int k = 0; k < DD; k += 4) {
    const int kb = k + 2 * g;
    v2f ah = *(const v2f*)(&lh[mn * LSTRIDE + kb]);
    v2f anr = *(const v2f*)(&ln[mn * LSTRIDE + kb]);
    v2f an; an.x = anr.x * invd; an.y = anr.y * invd;
    v2f bs; bs.x = Ws[(size_t)kb * DD + col0 + mn];
            bs.y = Ws[(size_t)(kb + 1) * DD + col0 + mn];
    v2f bn; bn.x = Wn[(size_t)kb * DD + col0 + mn];
            bn.y = Wn[(size_t)(kb + 1) * DD + col0 + mn];
    acc = __builtin_amdgcn_wmma_f32_16x16x4_f32(false, ah, false, bs, (short)0, acc, false, false);
    acc = __builtin_amdgcn_wmma_f32_16x16x4_f32(false, an, false, bn, (short)0, acc, false, false);
  }

  const float bb = bias[col0 + mn];
  #pragma unroll
  for (int r = 0; r < 8; ++r) {
    int row = row0 + r + 8 * g;      // C/D layout: VGPR r -> M = r + 8*(lane/16)
    float v = acc[r] + bb;
    if (relu) v = fmaxf(v, 0.0f);
    if (row < N) out[(size_t)row * DD + col0 + mn] = v;
  }
}

// ---------------------------------------------------------------- driver
extern "C" void kernel_launch(void* const* d_in, const int* in_sizes, int n_in,
                              void* d_out, int out_size, void* d_ws, size_t ws_size,
                              hipStream_t stream) {
  const float* feats = (const float*)d_in[0];
  const float* ew    = (const float*)d_in[1];
  const float* Ws0   = (const float*)d_in[2];
  const float* Wn0   = (const float*)d_in[3];
  const float* b0    = (const float*)d_in[4];
  const float* Ws1   = (const float*)d_in[5];
  const float* Wn1   = (const float*)d_in[6];
  const float* b1    = (const float*)d_in[7];
  const float* Ws2   = (const float*)d_in[8];
  const float* Wn2   = (const float*)d_in[9];
  const float* b2    = (const float*)d_in[10];
  const int*   src   = (const int*)d_in[11];
  const int*   dst   = (const int*)d_in[12];
  float* out = (float*)d_out;

  const int N = in_sizes[0] / DD;     // 50000
  const int E = in_sizes[1];          // 850000

  float* ws       = (float*)d_ws;
  float* deg_src  = ws;                         // N
  float* deg_dst  = ws + (size_t)N;             // N
  float* in_cnt   = ws + 2 * (size_t)N;         // N
  float* inv_deg  = ws + 3 * (size_t)N;         // N
  float* wnorm    = ws + 4 * (size_t)N;         // E
  float* neigh    = ws + 4 * (size_t)N + E;     // N*DD
  float* hA       = neigh + (size_t)N * DD;     // N*DD
  float* hB       = hA + (size_t)N * DD;        // N*DD

  const int T = 256;
  const int gE  = (E + T - 1) / T;
  const int gN  = (N + T - 1) / T;
  const int gAg = (E * 32 + T - 1) / T;         // wave per edge
  const int gMM = (N + 15) / 16;                // 16-row strips

  // degrees
  hipMemsetAsync(deg_src, 0, 3 * (size_t)N * sizeof(float), stream);
  sage_deg_kernel<<<gE, T, 0, stream>>>(src, dst, ew, deg_src, deg_dst, in_cnt, E);
  sage_invdeg_kernel<<<gN, T, 0, stream>>>(in_cnt, inv_deg, N);
  sage_edgenorm_kernel<<<gE, T, 0, stream>>>(src, dst, ew, deg_src, deg_dst, wnorm, E);

  // layer 0: feats -> hA
  hipMemsetAsync(neigh, 0, (size_t)N * DD * sizeof(float), stream);
  sage_aggregate_kernel<<<gAg, T, 0, stream>>>(src, dst, wnorm, feats, neigh, E);
  sage_gemm_kernel<<<gMM, T, 0, stream>>>(feats, neigh, inv_deg, Ws0, Wn0, b0, hA, N, 1);

  // layer 1: hA -> hB
  hipMemsetAsync(neigh, 0, (size_t)N * DD * sizeof(float), stream);
  sage_aggregate_kernel<<<gAg, T, 0, stream>>>(src, dst, wnorm, hA, neigh, E);
  sage_gemm_kernel<<<gMM, T, 0, stream>>>(hA, neigh, inv_deg, Ws1, Wn1, b1, hB, N, 1);

  // layer 2: hB -> out (no ReLU)
  hipMemsetAsync(neigh, 0, (size_t)N * DD * sizeof(float), stream);
  sage_aggregate_kernel<<<gAg, T, 0, stream>>>(src, dst, wnorm, hB, neigh, E);
  sage_gemm_kernel<<<gMM, T, 0, stream>>>(hB, neigh, inv_deg, Ws2, Wn2, b2, out, N, 0);
}